// YOLOLoss_32341103739086
// MI455X (gfx1250) — compile-verified
//
#include <hip/hip_runtime.h>
#include <hip/hip_bf16.h>

typedef float v2f __attribute__((ext_vector_type(2)));
typedef float v8f __attribute__((ext_vector_type(8)));

#define B_BOX 2
#define N_CLS 20
#define LAMBDA_NOOBJ 0.5f
#define EPS_IOU 1e-10f

// ---------------------------------------------------------------------------
// IoU of two center-format boxes (x, y, w, h) — matches reference ops exactly
// (w/2 == 0.5f*w exactly in f32).
// ---------------------------------------------------------------------------
__device__ __forceinline__ float iou_one(const float* b1, const float* b2) {
    float x1 = b1[0], y1 = b1[1], w1 = b1[2], h1 = b1[3];
    float x2 = b2[0], y2 = b2[1], w2 = b2[2], h2 = b2[3];
    float iw = fminf(x1 + 0.5f * w1, x2 + 0.5f * w2) -
               fmaxf(x1 - 0.5f * w1, x2 - 0.5f * w2);
    iw = fmaxf(0.0f, iw);
    float ih = fminf(y1 + 0.5f * h1, y2 + 0.5f * h2) -
               fmaxf(y1 - 0.5f * h1, y2 - 0.5f * h2);
    ih = fmaxf(0.0f, ih);
    float inter = iw * ih;
    float uni   = w1 * h1 + w2 * h2 - inter;
    return inter / (uni + EPS_IOU);
}

// ---------------------------------------------------------------------------
// Per-cell loss. p/t point at 30 consecutive floats held in registers
// (all indices are compile-time constants after inlining — no scratch).
// ---------------------------------------------------------------------------
__device__ __forceinline__ float cell_loss(const float* p, const float* t) {
    float iou0 = iou_one(p + 0, t + 0);
    float iou1 = iou_one(p + 5, t + 5);

    // argmax over 2 boxes; first-max-wins on ties (strict >)
    bool sel1    = (iou1 > iou0);
    bool has_obj = (t[4] > 0.0f);

    float px = sel1 ? p[5] : p[0];
    float py = sel1 ? p[6] : p[1];
    float tx = sel1 ? t[5] : t[0];
    float ty = sel1 ? t[6] : t[1];
    float dx = px - tx, dy = py - ty;
    float loss = has_obj ? (dx * dx + dy * dy) : 0.0f;

    // class SSE (unconditional) + argmax over target classes tracking p[gt]
    float bestT = t[10];
    float pcg   = p[10];
#pragma unroll
    for (int c = 1; c < N_CLS; ++c) {
        float tv = t[10 + c];
        float pv = p[10 + c];
        bool  gt = (tv > bestT);          // first-max-wins (strict >)
        bestT = gt ? tv : bestT;
        pcg   = gt ? pv : pcg;
    }
#pragma unroll
    for (int c = 0; c < N_CLS; ++c) {
        float d = p[10 + c] - t[10 + c];
        loss += d * d;
    }

    // confidence: diff = iou*pcg - iou ; weight 1 for selected box, 0.5 else
    float d0 = iou0 * pcg - iou0;
    float d1 = iou1 * pcg - iou1;
    float w0 = sel1 ? LAMBDA_NOOBJ : 1.0f;
    float w1 = sel1 ? 1.0f : LAMBDA_NOOBJ;
    loss += w0 * d0 * d0 + w1 * d1 * d1;
    return loss;
}

// ---------------------------------------------------------------------------
// Wave (32-lane) sum using V_WMMA_F32_16X16X4_F32:
//   A = ones(16x4), B(4x16) holds one partial per lane (other slots 0)
//   => D[0][n] = column sums; d[0] per lane covers all 16 columns across
//      lanes 0..15 (mirrored in 16..31). Four xor-shuffles finish the sum.
// Exact f32 arithmetic; EXEC must be all-ones at call site.
// ---------------------------------------------------------------------------
__device__ __forceinline__ float wave_sum_wmma(float x) {
    v2f a; a[0] = 1.0f; a[1] = 1.0f;   // A: all ones
    v2f b; b[0] = x;    b[1] = 0.0f;   // each lane's partial appears once in B
    v8f c = {};
    c = __builtin_amdgcn_wmma_f32_16x16x4_f32(
        /*neg_a=*/false, a, /*neg_b=*/false, b,
        /*c_mod=*/(short)0, c, /*reuse_a=*/false, /*reuse_b=*/false);
    float r = c[0];                    // pairwise (lane n, lane n+16) sums
    r += __shfl_xor(r, 1, 32);
    r += __shfl_xor(r, 2, 32);
    r += __shfl_xor(r, 4, 32);
    r += __shfl_xor(r, 8, 32);
    return r;                          // every lane holds the wave total
}

// ---------------------------------------------------------------------------
// Kernel 1: one thread per PAIR of cells (240 B per input -> 16B aligned,
// so the compiler can emit global_load_b128). Writes one partial per block.
// ---------------------------------------------------------------------------
__global__ __launch_bounds__(256) void yolo_partial_kernel(
    const float* __restrict__ preds, const float* __restrict__ targets,
    float* __restrict__ blk_out, long long ncells) {
    long long pair   = (long long)blockIdx.x * blockDim.x + threadIdx.x;
    long long npairs = (ncells + 1) >> 1;

    float loss = 0.0f;
    if (pair < npairs) {
        long long c0 = pair * 2;
        if (c0 + 1 < ncells) {
            // fast path: two cells, 15 x float4 loads per input
            const float4* p4 = reinterpret_cast<const float4*>(preds + c0 * 30);
            const float4* t4 = reinterpret_cast<const float4*>(targets + c0 * 30);
            float pf[60], tf[60];
#pragma unroll
            for (int i = 0; i < 15; ++i) {
                float4 v = p4[i];
                pf[4 * i + 0] = v.x; pf[4 * i + 1] = v.y;
                pf[4 * i + 2] = v.z; pf[4 * i + 3] = v.w;
                float4 u = t4[i];
                tf[4 * i + 0] = u.x; tf[4 * i + 1] = u.y;
                tf[4 * i + 2] = u.z; tf[4 * i + 3] = u.w;
            }
            loss = cell_loss(pf, tf) + cell_loss(pf + 30, tf + 30);
        } else {
            // odd tail: single cell via float2 (cell start is 8B aligned)
            const float2* p2 = reinterpret_cast<const float2*>(preds + c0 * 30);
            const float2* t2 = reinterpret_cast<const float2*>(targets + c0 * 30);
            float pf[30], tf[30];
#pragma unroll
            for (int i = 0; i < 15; ++i) {
                float2 v = p2[i]; pf[2 * i] = v.x; pf[2 * i + 1] = v.y;
                float2 u = t2[i]; tf[2 * i] = u.x; tf[2 * i + 1] = u.y;
            }
            loss = cell_loss(pf, tf);
        }
    }
    // all lanes reconverged here -> EXEC all-ones for WMMA
    float wsum = wave_sum_wmma(loss);

    __shared__ float smem[8];
    int lane = threadIdx.x & 31;
    int wid  = threadIdx.x >> 5;
    if (lane == 0) smem[wid] = wsum;
    __syncthreads();
    if (threadIdx.x == 0) {
        float s = 0.0f;
#pragma unroll
        for (int i = 0; i < 8; ++i) s += smem[i];
        blk_out[blockIdx.x] = s;
    }
}

// ---------------------------------------------------------------------------
// Kernel 2: single block folds the per-block partials, scales by 1/N and
// OVERWRITES d_out[0] (deterministic; no atomics; no zero-init needed).
// ---------------------------------------------------------------------------
__global__ __launch_bounds__(256) void yolo_finalize_kernel(
    const float* __restrict__ blk, int nblocks, float* __restrict__ out,
    float invN) {
    float s = 0.0f;
    for (int i = threadIdx.x; i < nblocks; i += 256) s += blk[i];

    float wsum = wave_sum_wmma(s);   // reconverged after loop -> EXEC full

    __shared__ float smem[8];
    int lane = threadIdx.x & 31;
    int wid  = threadIdx.x >> 5;
    if (lane == 0) smem[wid] = wsum;
    __syncthreads();
    if (threadIdx.x == 0) {
        float tot = 0.0f;
#pragma unroll
        for (int i = 0; i < 8; ++i) tot += smem[i];
        out[0] = tot * invN;
    }
}

extern "C" void kernel_launch(void* const* d_in, const int* in_sizes, int n_in,
                              void* d_out, int out_size, void* d_ws, size_t ws_size,
                              hipStream_t stream) {
    const float* preds   = (const float*)d_in[0];
    const float* targets = (const float*)d_in[1];

    const int D = B_BOX * 5 + N_CLS;                 // 30 floats per cell
    long long total  = (long long)in_sizes[0];       // N*S*S*D
    long long ncells = total / D;                    // N*S*S
    long long N      = ncells / 49;                  // S = 7
    float invN       = 1.0f / (float)N;

    long long npairs = (ncells + 1) >> 1;
    int threads = 256;
    int nblocks = (int)((npairs + threads - 1) / threads);

    float* blk = (float*)d_ws;                       // nblocks floats of scratch

    yolo_partial_kernel<<<nblocks, threads, 0, stream>>>(preds, targets, blk, ncells);
    yolo_finalize_kernel<<<1, threads, 0, stream>>>(blk, nblocks, (float*)d_out, invN);
}